// KNN_48971217109121
// MI455X (gfx1250) — compile-verified
//
#include <hip/hip_runtime.h>

typedef float v2f __attribute__((ext_vector_type(2)));
typedef float v8f __attribute__((ext_vector_type(8)));

#define BATCH 4
#define CH    3
#define MQ    4096
#define NP    8192
#define TOPK  16
#define NCHUNK 4            // waves splitting the N dimension per query tile
#define TILES_PER_BLOCK 2   // query tiles (of 16 queries) per block
#define BLOCK 256
#define CHUNK_N (NP / NCHUNK)   // 2048
#define TILES_PER_BATCH (MQ / 16)

__global__ __launch_bounds__(BLOCK) void knn_wmma_kernel(
    const float* __restrict__ Q,      // [B][C][M]
    const float* __restrict__ P,      // [B][C][N]
    float* __restrict__ outPts,       // [B][C][M][K]
    long long* __restrict__ outIdx)   // [B][M][K]
{
    __shared__ float sd[BLOCK * TOPK];                  // per-lane sorted dists
    __shared__ int   si[BLOCK * TOPK];                  // per-lane sorted idxs
    __shared__ float md[TILES_PER_BLOCK * 16 * TOPK];   // per-query merge scratch
    __shared__ int   mi[TILES_PER_BLOCK * 16 * TOPK];

    const int tid  = threadIdx.x;
    const int wave = tid >> 5;
    const int lane = tid & 31;
    const int half = lane >> 4;     // 0: K=0,1 / rows 0..7   1: K=2,3 / rows 8..15
    const int l    = lane & 15;

    const int qt    = wave >> 2;                // query tile within block
    const int chunk = wave & (NCHUNK - 1);      // N chunk
    const int tile  = blockIdx.x * TILES_PER_BLOCK + qt;
    const int b     = tile / TILES_PER_BATCH;
    const int q0    = (tile % TILES_PER_BATCH) * 16;

    const float* Qb = Q + (size_t)b * CH * MQ;
    const float* Pb = P + (size_t)b * CH * NP;

    // ---- B operand: queries as 4x16, scaled by -2; K=3 row = ones ----
    const int   qcol = q0 + l;
    const float qv0  = Qb[0 * MQ + qcol];
    const float qv1  = Qb[1 * MQ + qcol];
    const float qv2  = Qb[2 * MQ + qcol];
    const float q2s  = qv0 * qv0 + qv1 * qv1 + qv2 * qv2;

    v2f Bm;
    Bm.x = half ? (-2.0f * qv2) : (-2.0f * qv0);
    Bm.y = half ? 1.0f          : (-2.0f * qv1);

    // accumulator preloaded with ||q||^2 -> WMMA emits full distances
    v8f Cacc;
#pragma unroll
    for (int r = 0; r < 8; ++r) Cacc[r] = q2s;

    // ---- per-lane top-k (sorted ascending, in registers) ----
    float best[TOPK];
    int   bidx[TOPK];
#pragma unroll
    for (int j = 0; j < TOPK; ++j) { best[j] = 3.402823466e38f; bidx[j] = 0; }

    const int n_begin = chunk * CHUNK_N;
#pragma unroll 2
    for (int n0 = n_begin; n0 < n_begin + CHUNK_N; n0 += 16) {
        // ---- A operand: 16-point tile x (p0,p1,p2,||p||^2) ----
        const int   prow = n0 + l;
        const float p0   = Pb[0 * NP + prow];
        const float p1   = Pb[1 * NP + prow];
        const float p2v  = Pb[2 * NP + prow];
        v2f Am;
        Am.x = half ? p2v : p0;
        Am.y = half ? (p0 * p0 + p1 * p1 + p2v * p2v) : p1;

        // D[m,n] = ||q||^2 + ||p||^2 - 2 q.p   (16x16 tile, one instruction)
        v8f D = __builtin_amdgcn_wmma_f32_16x16x4_f32(
            false, Am, false, Bm, (short)0, Cacc, false, false);

        // ---- tile-level pre-filter: single guard for all 8 candidates ----
        const float t01 = fminf(D[0], D[1]);
        const float t23 = fminf(D[2], D[3]);
        const float t45 = fminf(D[4], D[5]);
        const float t67 = fminf(D[6], D[7]);
        const float tmin = fminf(fminf(t01, t23), fminf(t45, t67));

        if (tmin < best[TOPK - 1]) {               // rare path
#pragma unroll
            for (int r = 0; r < 8; ++r) {
                const float dist = D[r];
                const int   pidx = n0 + half * 8 + r;
                if (dist < best[TOPK - 1]) {
                    // Branchless sorted insert:
                    //   best[j] = max(best[j-1], min(dist, best[j]))
                    //   bidx[j] selected with -1/0 masks (lowers to cndmask)
                    int m_cur = -1;                // mask(dist < best[15]) guaranteed
#pragma unroll
                    for (int j = TOPK - 1; j >= 1; --j) {
                        const int m_prev = -(int)(dist < best[j - 1]); // shift mask
                        const int keepnew = (pidx & m_cur) | (bidx[j] & ~m_cur);
                        bidx[j] = (bidx[j - 1] & m_prev) | (keepnew & ~m_prev);
                        best[j] = fmaxf(best[j - 1], fminf(dist, best[j]));
                        m_cur = m_prev;
                    }
                    bidx[0] = (pidx & m_cur) | (bidx[0] & ~m_cur);
                    best[0] = fminf(dist, best[0]);
                }
            }
        }
    }

    // ---- dump per-lane sorted lists to LDS ----
#pragma unroll
    for (int j = 0; j < TOPK; ++j) {
        sd[tid * TOPK + j] = best[j];
        si[tid * TOPK + j] = bidx[j];
    }
    __syncthreads();

    // ---- one thread per query: 7 sequential 2-way merges of 8 lists ----
    if (tid < TILES_PER_BLOCK * 16) {
        const int qt2   = tid >> 4;
        const int l2    = tid & 15;
        const int sbase = tid * TOPK;

        // seed scratch with list (chunk 0, half 0)
        {
            const int lb0 = ((qt2 * NCHUNK + 0) * 32 + l2) * TOPK;
#pragma unroll
            for (int j = 0; j < TOPK; ++j) { md[sbase + j] = sd[lb0 + j]; mi[sbase + j] = si[lb0 + j]; }
        }
        for (int s = 1; s < 2 * NCHUNK; ++s) {
            const int c  = s >> 1;
            const int hh = s & 1;
            const int lb = ((qt2 * NCHUNK + c) * 32 + hh * 16 + l2) * TOPK;
            float rm[TOPK];
            int   ri[TOPK];
            int ia = 0, ib = 0;
#pragma unroll
            for (int kk = 0; kk < TOPK; ++kk) {
                const float da = md[sbase + ia];
                const float db = sd[lb + ib];
                const int   ai = mi[sbase + ia];
                const int   bi = si[lb + ib];
                const int   m  = -(int)(da <= db);   // -1: take A, 0: take B
                rm[kk] = fminf(da, db);
                ri[kk] = (ai & m) | (bi & ~m);
                ia -= m;            // m=-1 -> ia++
                ib += 1 + m;        // m=0  -> ib++
            }
#pragma unroll
            for (int kk = 0; kk < TOPK; ++kk) { md[sbase + kk] = rm[kk]; mi[sbase + kk] = ri[kk]; }
        }

        // ---- write outputs: indices (i64) + gathered neighbor coords ----
        const int tile2 = blockIdx.x * TILES_PER_BLOCK + qt2;
        const int b2    = tile2 / TILES_PER_BATCH;
        const int m     = (tile2 % TILES_PER_BATCH) * 16 + l2;
        const float* Pb2 = P + (size_t)b2 * CH * NP;
#pragma unroll
        for (int kk = 0; kk < TOPK; ++kk) {
            const int idx = mi[sbase + kk];
            outIdx[((size_t)b2 * MQ + m) * TOPK + kk] = (long long)idx;
#pragma unroll
            for (int c2 = 0; c2 < CH; ++c2) {
                outPts[(((size_t)b2 * CH + c2) * MQ + m) * TOPK + kk] = Pb2[c2 * NP + idx];
            }
        }
    }
}

extern "C" void kernel_launch(void* const* d_in, const int* in_sizes, int n_in,
                              void* d_out, int out_size, void* d_ws, size_t ws_size,
                              hipStream_t stream) {
    // setup_inputs order: k (int scalar), query (f32 B*C*M), points (f32 B*C*N)
    const float* Q = (const float*)d_in[1];
    const float* P = (const float*)d_in[2];

    float*     outPts = (float*)d_out;                                         // B*C*M*K f32
    long long* outIdx = (long long*)(outPts + (size_t)BATCH * CH * MQ * TOPK); // B*M*K i64

    const int tiles = BATCH * TILES_PER_BATCH;       // 1024 query tiles
    const int grid  = tiles / TILES_PER_BLOCK;       // 512 blocks x 256 threads
    knn_wmma_kernel<<<grid, BLOCK, 0, stream>>>(Q, P, outPts, outIdx);
}